// LocalBatchInstanceNormalization_42580305773122
// MI455X (gfx1250) — compile-verified
//
#include <hip/hip_runtime.h>

// LocalBatchInstanceNormalization for MI455X (gfx1250).
//
// x: (32,128,128,128) NHWC fp32.  Memory floor ~1 GB of traffic (two mandatory
// stat passes + one fused read/write pass) => ~44us @ 23.3 TB/s.  Everything
// else (box filters, normalization) is fused into one sliding-row pipeline
// held in CDNA5's 320KB LDS, with the horizontal 6-tap SAME box filter done
// as a banded-matrix matmul on V_WMMA_F32_16X16X4_F32 (0/1 coefficients are
// exact in fp32, K-chunks outside the band contribute zero).
//
// Round-1 fix: uniform compile-time trip count (6) for the kc loop so the
// compiler cannot EXEC-mask lanes around the WMMA (ISA: EXEC must be all 1s).
// Round-2 polish: prefetch with high locality (near-scope instead of SYS),
// and an interior fast path for the vertical 6-row sums (no per-row branch
// checks for the 123/128 unclipped rows).

#define EPS    1e-5f
#define WIDTH  128
#define HEIGHT 128
#define CHAN   128
#define BATCH  32
#define CG     16            // channels per workgroup
#define RS     130           // padded LDS row stride per channel (bank-safe, 8B aligned)
#define ROWF   (CG * RS)     // floats per buffered row (2080)
#define NPIX   (BATCH * HEIGHT * WIDTH)   // pixels per channel = 524288

#define SH_FLOATS (29 * ROWF + 208)
#define SH_BYTES  (SH_FLOATS * 4)

typedef float v2f __attribute__((ext_vector_type(2)));
typedef float v8f __attribute__((ext_vector_type(8)));

// Horizontal 6-tap SAME box sum over one buffered row (16 channels x 128 w),
// stored channel-major in LDS (c*RS + w).  HX = X(16x128) * Band(128x128),
// Band[k][w] = 1 iff w-2 <= k <= w+3.  One wave per 16-wide output tile
// (8 waves = full row); exactly 6 K-chunks issued per wave, with chunks that
// fall outside [0,31] neutralized via zero band coefficients.
// Fragment layouts per CDNA5 ISA 7.12.2:
//   A 16x4 f32 : lanes 0-15 hold M=lane {K0,K1}; lanes 16-31 hold {K2,K3}
//   B 4x16 f32 : vgpr j, half h -> K = j + 2h, N = lane&15
//   D 16x16 f32: vgpr r, half h -> M = r + 8h, N = lane&15
__device__ inline void hbox16(const float* __restrict__ rowIn,
                              float* __restrict__ rowOut) {
  const int lane = threadIdx.x & 31;
  const int wave = threadIdx.x >> 5;   // output tile index (w = 16*wave .. +15)
  const int half = lane >> 4;
  const int l15  = lane & 15;
  const int wn   = wave * 16 + l15;    // this lane's output column (N dim)
  const float* aRow = rowIn + l15 * RS;

  v8f acc = {};
#pragma unroll
  for (int i = 0; i < 6; ++i) {
    const int kc   = wave * 4 - 1 + i;               // may be -1 or 32 at edges
    const bool ok  = (kc >= 0) && (kc <= 31);
    const int kcc  = (kc < 0) ? 0 : ((kc > 31) ? 31 : kc);
    const int k0   = kcc * 4 + half * 2;             // clamped, in-bounds A addr
    const int kk0  = kc * 4 + half * 2;              // logical K for band test
    v2f a = *(const v2f*)(aRow + k0);
    v2f b;
    b.x = (ok && kk0     >= wn - 2 && kk0     <= wn + 3) ? 1.0f : 0.0f;
    b.y = (ok && kk0 + 1 >= wn - 2 && kk0 + 1 <= wn + 3) ? 1.0f : 0.0f;
    acc = __builtin_amdgcn_wmma_f32_16x16x4_f32(false, a, false, b,
                                                (short)0, acc, false, false);
  }
#pragma unroll
  for (int r = 0; r < 8; ++r) {
    rowOut[(r + half * 8) * RS + wave * 16 + l15] = acc[r];
  }
}

// Vertical 6-row sum from an 8-deep ring buffer.
// Fast path: all rows r-2..r+3 known in-range (uniform interior condition).
__device__ inline float vsum6_fast(const float* __restrict__ ring, int r, int a) {
  float s = ring[((r - 2) & 7) * ROWF + a];
  s += ring[((r - 1) & 7) * ROWF + a];
  s += ring[((r + 0) & 7) * ROWF + a];
  s += ring[((r + 1) & 7) * ROWF + a];
  s += ring[((r + 2) & 7) * ROWF + a];
  s += ring[((r + 3) & 7) * ROWF + a];
  return s;
}

// Edge path: rows outside [0,HEIGHT) contribute 0.
__device__ inline float vsum6_edge(const float* __restrict__ ring, int r, int a) {
  float s = 0.0f;
#pragma unroll
  for (int d = -2; d <= 3; ++d) {
    const int rr = r + d;
    if (rr >= 0 && rr < HEIGHT) s += ring[(rr & 7) * ROWF + a];
  }
  return s;
}

// ---------------- batch statistics ----------------

__global__ __launch_bounds__(256) void k_zero(float* acc) {
  acc[threadIdx.x] = 0.0f;
}

__global__ __launch_bounds__(256) void k_sum(const float* __restrict__ x,
                                             float* __restrict__ acc) {
  __shared__ float red[256];
  const int tid = threadIdx.x;
  const int c   = tid & 127;
  const size_t p0 = (size_t)blockIdx.x * 512 + (tid >> 7);
  float s = 0.0f;
  for (int i = 0; i < 256; ++i)
    s += x[(p0 + (size_t)2 * i) * CHAN + c];
  red[tid] = s;
  __syncthreads();
  if (tid < 128) atomicAdd(&acc[c], red[tid] + red[tid + 128]);
}

__global__ __launch_bounds__(256) void k_sumabs(const float* __restrict__ x,
                                                float* __restrict__ acc) {
  __shared__ float red[256];
  const int tid = threadIdx.x;
  const int c   = tid & 127;
  const float mean = acc[c] * (1.0f / (float)NPIX);
  const size_t p0 = (size_t)blockIdx.x * 512 + (tid >> 7);
  float s = 0.0f;
  for (int i = 0; i < 256; ++i)
    s += fabsf(x[(p0 + (size_t)2 * i) * CHAN + c] - mean);
  red[tid] = s;
  __syncthreads();
  if (tid < 128) atomicAdd(&acc[128 + c], red[tid] + red[tid + 128]);
}

// ---------------- fused main pass ----------------
// One workgroup = one (n, 16-channel group).  Sliding pipeline over h:
//   step s: load x row s -> WMMA hbox -> RHX[s]
//           r = s-3: lm[r] = vsum6(RHX)/cnt; t[r]=|x-lm|; WMMA hbox -> RHT[r]
//           ho = s-6: mad = vsum6(RHT)/cnt; emit output row ho.
__global__ __launch_bounds__(256) void k_main(const float* __restrict__ x,
                                              const float* __restrict__ acc,
                                              const float* __restrict__ gamma,
                                              const float* __restrict__ beta,
                                              const float* __restrict__ lbw,
                                              float* __restrict__ out) {
  extern __shared__ float lds[];
  float* RX  = lds;               // 8-row ring: raw x
  float* RHX = lds + 8  * ROWF;   // 8-row ring: hbox(x)
  float* RHT = lds + 16 * ROWF;   // 8-row ring: hbox(t)
  float* RLM = lds + 24 * ROWF;   // 4-row ring: local mean
  float* TR  = lds + 28 * ROWF;   // scratch row: t = |x - lm|
  float* ICW = lds + 29 * ROWF;   // [128] 1/countW
  float* TAB = ICW + 128;         // [0:16)mean [16:32)1/(sig+eps) [32:48)w [48:64)gamma [64:80)beta

  const int tid = threadIdx.x;
  const int n   = blockIdx.x >> 3;
  const int cg  = blockIdx.x & 7;
  const float* xb = x   + (size_t)n * HEIGHT * WIDTH * CHAN + cg * CG;
  float*       ob = out + (size_t)n * HEIGHT * WIDTH * CHAN + cg * CG;

  if (tid < 128) {
    const int w  = tid;
    const int lo = (w - 2 < 0) ? 0 : w - 2;
    const int hi = (w + 3 > WIDTH - 1) ? WIDTH - 1 : w + 3;
    ICW[w] = 1.0f / (float)(hi - lo + 1);
  }
  if (tid < 16) {
    const int c = cg * CG + tid;
    const float mean = acc[c]       * (1.0f / (float)NPIX);
    const float sig  = acc[128 + c] * (1.0f / (float)NPIX);
    TAB[tid]      = mean;
    TAB[16 + tid] = 1.0f / (sig + EPS);
    TAB[32 + tid] = lbw[c];
    TAB[48 + tid] = gamma[c];
    TAB[64 + tid] = beta[c];
  }
  __syncthreads();

  for (int s = 0; s < HEIGHT + 6; ++s) {
    // ---- stage 1: load x row s (coalesced float4), hbox -> RHX ----
    if (s < HEIGHT) {
      const float* rowg = xb + (size_t)s * WIDTH * CHAN;
      float* rx = RX + (s & 7) * ROWF;
#pragma unroll
      for (int k = 0; k < 2; ++k) {
        const int j  = tid + k * 256;        // 512 float4 chunks per row
        const int w  = j >> 2;
        const int c4 = (j & 3) * 4;
        const float4 v = *(const float4*)(rowg + (size_t)w * CHAN + c4);
        rx[(c4 + 0) * RS + w] = v.x;
        rx[(c4 + 1) * RS + w] = v.y;
        rx[(c4 + 2) * RS + w] = v.z;
        rx[(c4 + 3) * RS + w] = v.w;
      }
      if (s + 1 < HEIGHT)   // prefetch next row, near scope (high locality)
        __builtin_prefetch(rowg + (size_t)WIDTH * CHAN + tid * 64, 0, 3);
      __syncthreads();
      hbox16(rx, RHX + (s & 7) * ROWF);
    }
    __syncthreads();

    // ---- stage 2: local mean row r, t row, hbox(t) -> RHT ----
    const int r = s - 3;
    if (r >= 0 && r < HEIGHT) {
      const int lo = (r - 2 < 0) ? 0 : r - 2;
      const int hi = (r + 3 > HEIGHT - 1) ? HEIGHT - 1 : r + 3;
      const float ich = 1.0f / (float)(hi - lo + 1);
      const bool interior = (r >= 2) && (r <= HEIGHT - 4);   // uniform
      float* lm = RLM + (r & 3) * ROWF;
      const float* rx = RX + (r & 7) * ROWF;
#pragma unroll
      for (int k = 0; k < 8; ++k) {
        const int idx = tid + k * 256;       // 2048 elems per row
        const int c = idx & 15, w = idx >> 4;
        const int a = c * RS + w;
        const float sum = interior ? vsum6_fast(RHX, r, a)
                                   : vsum6_edge(RHX, r, a);
        const float v = sum * (ICW[w] * ich);
        lm[a] = v;
        TR[a] = fabsf(rx[a] - v);
      }
      __syncthreads();
      hbox16(TR, RHT + (r & 7) * ROWF);
    }
    __syncthreads();

    // ---- stage 3: local MAD + blend, emit output row ho ----
    const int ho = s - 6;
    if (ho >= 0) {
      const int lo = (ho - 2 < 0) ? 0 : ho - 2;
      const int hi = (ho + 3 > HEIGHT - 1) ? HEIGHT - 1 : ho + 3;
      const float ich = 1.0f / (float)(hi - lo + 1);
      const bool interior = (ho >= 2) && (ho <= HEIGHT - 4);   // uniform
      const float* rx = RX  + (ho & 7) * ROWF;
      const float* lm = RLM + (ho & 3) * ROWF;
      float* og = ob + (size_t)ho * WIDTH * CHAN;
#pragma unroll
      for (int k = 0; k < 8; ++k) {
        const int idx = tid + k * 256;
        const int c = idx & 15, w = idx >> 4;
        const int a = c * RS + w;
        const float sum = interior ? vsum6_fast(RHT, ho, a)
                                   : vsum6_edge(RHT, ho, a);
        const float mad = sum * (ICW[w] * ich);
        const float xv  = rx[a];
        const float xl  = (xv - lm[a]) / (mad + EPS);
        const float xbt = (xv - TAB[c]) * TAB[16 + c];
        const float wb  = TAB[32 + c];
        og[(size_t)w * CHAN + c] =
            (wb * xl + (1.0f - wb) * xbt) * TAB[48 + c] + TAB[64 + c];
      }
    }
    __syncthreads();
  }
}

extern "C" void kernel_launch(void* const* d_in, const int* in_sizes, int n_in,
                              void* d_out, int out_size, void* d_ws, size_t ws_size,
                              hipStream_t stream) {
  (void)in_sizes; (void)n_in; (void)out_size; (void)ws_size;
  const float* x     = (const float*)d_in[0];
  const float* gamma = (const float*)d_in[1];
  const float* beta  = (const float*)d_in[2];
  const float* lbw   = (const float*)d_in[3];
  float* out = (float*)d_out;
  float* acc = (float*)d_ws;   // [0:128) channel sums, [128:256) sum|x-mean|

  // 242KB dynamic LDS (> default 64KB cap); idempotent, deterministic.
  (void)hipFuncSetAttribute(reinterpret_cast<const void*>(k_main),
                            hipFuncAttributeMaxDynamicSharedMemorySize,
                            (int)SH_BYTES);

  k_zero  <<<1,    256, 0, stream>>>(acc);
  k_sum   <<<1024, 256, 0, stream>>>(x, acc);
  k_sumabs<<<1024, 256, 0, stream>>>(x, acc);
  k_main  <<<dim3(BATCH * 8), 256, SH_BYTES, stream>>>(x, acc, gamma, beta,
                                                       lbw, out);
}